// LookHopsPool_25125558681612
// MI455X (gfx1250) — compile-verified
//
#include <hip/hip_runtime.h>
#include <hip/hip_bf16.h>

// ---------------- problem constants (from the reference) ----------------
#define BGRAPH   64
#define N_PER    2048
#define F_DIM    64
#define NNODES   (BGRAPH * N_PER)        // 131072
#define NEDGE    1048576
#define A_DIM    8
#define K_SEL    1639                    // ceil(0.8 * 2048)
#define P_SEL    (BGRAPH * K_SEL)        // 104896
#define KTH_RANK 838861                  // 1-based rank of the threshold value

// output regions (float elements), in reference return order
#define O_XOUT   ((size_t)0)                          // P_SEL*64
#define O_EIDX   ((size_t)P_SEL * F_DIM)              // 6,713,344
#define O_EATTR  (O_EIDX  + (size_t)2 * NEDGE)        // 8,810,496
#define O_BATCH  (O_EATTR + (size_t)A_DIM * NEDGE)    // 17,199,104
#define O_SEL    (O_BATCH + (size_t)P_SEL)            // 17,304,000

// ---------------- async global->LDS availability ------------------------
#if defined(__has_builtin)
#if __has_builtin(__builtin_amdgcn_global_load_async_to_lds_b128) && \
    __has_builtin(__builtin_amdgcn_s_wait_asynccnt)
#define HAVE_ASYNC_LDS 1
#endif
#endif
#ifndef HAVE_ASYNC_LDS
#define HAVE_ASYNC_LDS 0
#endif

// ---------------- WMMA / async types ----------------
typedef __attribute__((ext_vector_type(2))) float v2f;
typedef __attribute__((ext_vector_type(8))) float v8f;
typedef __attribute__((ext_vector_type(4))) int   v4i;
typedef __attribute__((address_space(1))) v4i     g_v4i;   // global int4 vec
typedef __attribute__((address_space(3))) v4i     l_v4i;   // LDS int4 vec
union V8F { v8f v; float f[8]; };

// ---------------- helpers ----------------
__device__ __forceinline__ unsigned ord_desc(float f) {
  unsigned u = __float_as_uint(f);
  u = (u & 0x80000000u) ? ~u : (u | 0x80000000u);   // ascending-orderable
  return ~u;                                        // descending-orderable
}

// ---------------- kernel 1: init new_idx / select-state ----------------
__global__ void init_kernel(int* new_idx, unsigned* state) {
  int i = blockIdx.x * blockDim.x + threadIdx.x;
  if (i < NNODES) new_idx[i] = -1;
  if (i == 0) { state[0] = 0u; state[1] = (unsigned)KTH_RANK; }
}

// ---------------- kernel 2: per-graph top-k (bitonic sort in LDS) -------
__global__ void __launch_bounds__(1024)
topk_kernel(const float* __restrict__ x_score, int* __restrict__ perm,
            int* __restrict__ new_idx) {
  __shared__ unsigned long long keys[N_PER];
  const int g   = blockIdx.x;
  const int tid = threadIdx.x;                  // 0..1023
  for (int i = tid; i < N_PER; i += 1024) {
    float s = x_score[(size_t)g * N_PER + i];
    keys[i] = ((unsigned long long)ord_desc(s) << 32) | (unsigned)i;
  }
  for (unsigned k = 2; k <= N_PER; k <<= 1) {
    for (unsigned j = k >> 1; j > 0; j >>= 1) {
      __syncthreads();
      unsigned i = 2u * tid - (tid & (j - 1u));
      unsigned long long a = keys[i], b = keys[i + j];
      bool up = ((i & k) == 0);
      if ((a > b) == up) { keys[i] = b; keys[i + j] = a; }
    }
  }
  __syncthreads();
  for (int i = tid; i < K_SEL; i += 1024) {
    int idx  = (int)(keys[i] & 0xFFFFFFFFull);
    int node = g * N_PER + idx;
    int p    = g * K_SEL + i;
    perm[p] = node;
    new_idx[node] = p;
  }
}

// ---------------- kernel 3: per-node squared norms ----------------------
__global__ void sqnorm_kernel(const float* __restrict__ x, float* __restrict__ sqn) {
  const int t    = threadIdx.x;
  const int row  = blockIdx.x * 16 + (t >> 4);   // 16 rows per block
  const int c4   = (t & 15) * 4;
  const float4 v = *(const float4*)(x + (size_t)row * F_DIM + c4);
  float s = v.x * v.x + v.y * v.y + v.z * v.z + v.w * v.w;
  s += __shfl_xor(s, 8, 16);
  s += __shfl_xor(s, 4, 16);
  s += __shfl_xor(s, 2, 16);
  s += __shfl_xor(s, 1, 16);
  if ((t & 15) == 0) sqn[row] = s;
}

// ---------------- kernel 4: gather x_out and batch_out ------------------
__global__ void gather_kernel(const float* __restrict__ x, const int* __restrict__ perm,
                              const int* __restrict__ batch, float* __restrict__ out) {
  const size_t tid = (size_t)blockIdx.x * blockDim.x + threadIdx.x;
  if (tid >= (size_t)P_SEL * 16) return;
  const int p  = (int)(tid >> 4);
  const int q4 = (int)(tid & 15) * 4;
  const int node = perm[p];
  __builtin_prefetch(x + (size_t)node * F_DIM, 0, 0);   // global_prefetch_b8
  const float4 v = *(const float4*)(x + (size_t)node * F_DIM + q4);
  *(float4*)(out + O_XOUT + (size_t)p * F_DIM + q4) = v;
  if (q4 == 0) out[O_BATCH + p] = (float)batch[node];
}

// ---------------- kernel 5: filter_adj (remap + mask attrs) -------------
__global__ void edge_filter_kernel(const int* __restrict__ erow, const int* __restrict__ ecol,
                                   const float* __restrict__ eattr,
                                   const int* __restrict__ new_idx,
                                   float* __restrict__ out) {
  const size_t e = (size_t)blockIdx.x * blockDim.x + threadIdx.x;
  if (e >= NEDGE) return;
  const int r = new_idx[erow[e]];
  const int c = new_idx[ecol[e]];
  const bool valid = (r >= 0) & (c >= 0);
  out[O_EIDX + e]         = (float)(valid ? r : -1);
  out[O_EIDX + NEDGE + e] = (float)(valid ? c : -1);
  float4 a0 = *(const float4*)(eattr + e * A_DIM);
  float4 a1 = *(const float4*)(eattr + e * A_DIM + 4);
  if (!valid) { a0 = make_float4(0.f, 0.f, 0.f, 0.f); a1 = a0; }
  *(float4*)(out + O_EATTR + e * A_DIM)     = a0;
  *(float4*)(out + O_EATTR + e * A_DIM + 4) = a1;
}

// ---------------- kernel 6: edge scores via f32 WMMA Gram diagonal ------
// S[e] = exp(||x[row]-x[col]||) = exp(sqrt(nr + nc - 2*dot))
// 4 waves/block, 16 edges/wave.  Rows are async-copied global->LDS (f32,
// per-lane addresses, ASYNCcnt), then 16x V_WMMA_F32_16X16X4_F32 cover K=64.
__global__ void __launch_bounds__(128)
edge_s_kernel(const float* __restrict__ x, const int* __restrict__ erow,
              const int* __restrict__ ecol, const float* __restrict__ sqn,
              float* __restrict__ S) {
  __shared__ float lsA[4][16][F_DIM];   // A rows = x[row]  (16 KB)
  __shared__ float lsB[4][16][F_DIM];   // B cols = x[col]  (16 KB)
  __shared__ float lsN[4][32];
  const int w    = threadIdx.x >> 5;
  const int lane = threadIdx.x & 31;
  const int e0   = blockIdx.x * 64 + w * 16;
  const int slot = lane & 15;

  // stage: lanes 0-15 fetch x[row], lanes 16-31 fetch x[col] (256 B each)
  const int node = (lane < 16) ? erow[e0 + slot] : ecol[e0 + slot];
  const float* src = x + (size_t)node * F_DIM;
  float* dst = (lane < 16) ? lsA[w][slot] : lsB[w][slot];
#if HAVE_ASYNC_LDS
  #pragma unroll
  for (int j = 0; j < F_DIM; j += 4) {
    __builtin_amdgcn_global_load_async_to_lds_b128(
        (g_v4i*)(float*)(src + j), (l_v4i*)(dst + j), 0, 0);
  }
#else
  #pragma unroll
  for (int j = 0; j < F_DIM; j += 4)
    *(float4*)(dst + j) = *(const float4*)(src + j);
#endif
  lsN[w][lane] = sqn[node];
#if HAVE_ASYNC_LDS
  __builtin_amdgcn_s_wait_asynccnt(0);
#endif
  __syncthreads();

  // f32 WMMA 16x16x4: A 16x4 -> lane m: K{0,1} (lanes 0-15) / K{2,3} (16-31)
  //                   B 4x16 -> lane n: VGPR0 K=0(lo)/2(hi), VGPR1 K=1/3
  const int m   = lane & 15;
  const int khi = (lane >= 16) ? 2 : 0;
  v8f c = {};
  #pragma unroll
  for (int k = 0; k < F_DIM; k += 4) {
    v2f a, b;
    a.x = lsA[w][m][k + khi];
    a.y = lsA[w][m][k + khi + 1];
    b.x = lsB[w][m][k + khi];
    b.y = lsB[w][m][k + khi + 1];
    c = __builtin_amdgcn_wmma_f32_16x16x4_f32(false, a, false, b,
                                              (short)0, c, false, false);
  }

  // diagonal: D(l,l) -> lane l, elem l (l<8); lane l+16, elem l-8 (l>=8)
  V8F cu; cu.v = c;
  const int g = (lane < 8) ? lane : ((lane >= 24) ? (lane - 24) : -1);
  float dot = 0.f;
  #pragma unroll
  for (int i = 0; i < 8; ++i)
    if (g == i) dot = cu.f[i];          // cndmask chain, no scratch spill
  if (g >= 0) {
    const int l = (lane < 8) ? lane : (lane - 16);
    float sq = lsN[w][l] + lsN[w][16 + l] - 2.0f * dot;
    S[e0 + l] = (sq > 0.0f) ? expf(sqrtf(sq)) : 1.0f;
  }
}

// ---------------- kernels 7-9: radix select of the KTH-largest ----------
__global__ void zero_hist_kernel(unsigned* hist) { hist[threadIdx.x] = 0u; }

__global__ void hist_kernel(const float* __restrict__ S, unsigned* __restrict__ hist,
                            const unsigned* __restrict__ state, int shift) {
  __shared__ unsigned lh[256];
  lh[threadIdx.x] = 0u;
  __syncthreads();
  const unsigned prefix = state[0];
  const size_t stride = (size_t)gridDim.x * blockDim.x;
  for (size_t e = (size_t)blockIdx.x * blockDim.x + threadIdx.x; e < NEDGE; e += stride) {
    unsigned u = __float_as_uint(S[e]);             // S > 0: uint order == float order
    bool ok = (shift == 24) || ((u >> (shift + 8)) == prefix);
    if (ok) atomicAdd(&lh[(u >> shift) & 255u], 1u);
  }
  __syncthreads();
  atomicAdd(&hist[threadIdx.x], lh[threadIdx.x]);
}

__global__ void scan_kernel(const unsigned* __restrict__ hist, unsigned* __restrict__ state) {
  unsigned k = state[1];
  unsigned cum = 0, bin = 0;
  for (int b = 255; b >= 0; --b) {
    unsigned c = hist[b];
    if (cum + c >= k) { bin = (unsigned)b; k -= cum; break; }
    cum += c;
  }
  state[0] = (state[0] << 8) | bin;
  state[1] = k;
}

__global__ void select_kernel(const float* __restrict__ S, const unsigned* __restrict__ state,
                              float* __restrict__ out) {
  const size_t e = (size_t)blockIdx.x * blockDim.x + threadIdx.x;
  if (e >= NEDGE) return;
  const float th = __uint_as_float(state[0]);
  out[O_SEL + e] = (S[e] > th) ? 1.0f : 0.0f;
}

// ---------------- host-side launch --------------------------------------
extern "C" void kernel_launch(void* const* d_in, const int* in_sizes, int n_in,
                              void* d_out, int out_size, void* d_ws, size_t ws_size,
                              hipStream_t stream) {
  const float* x       = (const float*)d_in[0];
  const float* x_score = (const float*)d_in[1];
  const int*   eidx    = (const int*)d_in[2];
  const float* eattr   = (const float*)d_in[3];
  const int*   batch   = (const int*)d_in[4];
  const int*   erow    = eidx;
  const int*   ecol    = eidx + NEDGE;
  float* out = (float*)d_out;

  char* ws = (char*)d_ws;
  int*      perm    = (int*)      (ws + 0);         // P_SEL      ints  (419,584 B)
  int*      new_idx = (int*)      (ws + 419584);    // NNODES     ints  (524,288 B)
  float*    sqn     = (float*)    (ws + 943872);    // NNODES     f32   (524,288 B)
  float*    S       = (float*)    (ws + 1468160);   // NEDGE      f32   (4 MB)
  unsigned* hist    = (unsigned*) (ws + 5662464);   // 256        u32
  unsigned* state   = (unsigned*) (ws + 5663488);   // 4          u32

  init_kernel<<<(NNODES + 255) / 256, 256, 0, stream>>>(new_idx, state);
  topk_kernel<<<BGRAPH, 1024, 0, stream>>>(x_score, perm, new_idx);
  sqnorm_kernel<<<NNODES / 16, 256, 0, stream>>>(x, sqn);
  gather_kernel<<<((size_t)P_SEL * 16 + 255) / 256, 256, 0, stream>>>(x, perm, batch, out);
  edge_filter_kernel<<<NEDGE / 256, 256, 0, stream>>>(erow, ecol, eattr, new_idx, out);
  edge_s_kernel<<<NEDGE / 64, 128, 0, stream>>>(x, erow, ecol, sqn, S);
  for (int shift = 24; shift >= 0; shift -= 8) {
    zero_hist_kernel<<<1, 256, 0, stream>>>(hist);
    hist_kernel<<<1024, 256, 0, stream>>>(S, hist, state, shift);
    scan_kernel<<<1, 1, 0, stream>>>(hist, state);
  }
  select_kernel<<<NEDGE / 256, 256, 0, stream>>>(S, state, out);
}